// EmbeddingLayer_35751307772507
// MI455X (gfx1250) — compile-verified
//
#include <hip/hip_runtime.h>

typedef __attribute__((ext_vector_type(2))) float v2f;
typedef __attribute__((ext_vector_type(8))) float v8f;

// out[bt, h, f] = x[bt, f] * W[f, h] + b[f, h]
//   B*T = num_bt, F = 32, H = 64.
// Per (b,t):  D(64x32) = Wt(64x32) * diag(x(32)) + Bt(64x32)
// computed as 8 independent 16x16 tiles (one per wave in a 256-thread block),
// each tile = chain of 4x V_WMMA_F32_16X16X4_F32 with diagonal B operands.
__global__ __launch_bounds__(256) void fused_mul_bias_wmma(
    const float* __restrict__ x,
    const float* __restrict__ W,     // [F=32, H=64]
    const float* __restrict__ bias,  // [F=32, H=64]
    float* __restrict__ out,         // [num_bt, H=64, F=32]
    int num_bt)
{
  const int lane = threadIdx.x & 31;
  const int wave = threadIdx.x >> 5;   // 0..7
  const int h0 = (wave & 3) << 4;      // 0,16,32,48
  const int f0 = (wave >> 2) << 4;     // 0,16
  const int n  = lane & 15;            // tile column (f offset) for B/C/D
  const int hi = lane >> 4;            // lane half

  // ---- A operands: Wt tile, 16(h) x 4(f) per k-step. Loop-invariant. ----
  // A 16x4 f32 layout (ISA 7.12.2): lane L -> row M = L&15;
  //   VGPR0 holds K = 2*(L>>4), VGPR1 holds K = 2*(L>>4)+1.
  // A_k[m, j] = Wt[h0+m, f0+4k+j] = W[f0+4k+j, h0+m]
  v2f A[4];
#pragma unroll
  for (int k = 0; k < 4; ++k) {
    const int kb = f0 + 4 * k + 2 * hi;
    A[k].x = W[(kb + 0) * 64 + h0 + n];
    A[k].y = W[(kb + 1) * 64 + h0 + n];
  }

  // ---- C operand: bias^T tile in C/D layout. Loop-invariant. ----
  // D layout: VGPR v, lane L -> row M = v + 8*(L>>4), col N = L&15.
  v8f Cb;
#pragma unroll
  for (int v = 0; v < 8; ++v)
    Cb[v] = bias[(f0 + n) * 64 + (h0 + v + 8 * hi)];

  // ---- Per-lane diag placement masks for the B operands. ----
  // B 4x16 f32 layout (mirrors A's K striping): lane L -> col N = L&15;
  //   VGPR0 holds row j = 2*(L>>4), VGPR1 holds row j = 2*(L>>4)+1.
  // B_k[j, N] = x[f0+N] iff N == 4k + j, else 0.
  bool m0[4], m1[4];
#pragma unroll
  for (int k = 0; k < 4; ++k) {
    m0[k] = (n == 4 * k + 2 * hi);
    m1[k] = (n == 4 * k + 2 * hi + 1);
  }

  for (int bt = blockIdx.x; bt < num_bt; bt += gridDim.x) {
    // Coalesced x row load (lanes 0-15 / 16-31 each cover one 64B segment).
    const float xv = x[bt * 32 + f0 + n];

    // Speculative prefetch of the next iteration's x row.
    __builtin_prefetch(x + (size_t)(bt + gridDim.x) * 32 + f0 + n, 0, 0);

    v8f D = Cb;
#pragma unroll
    for (int k = 0; k < 4; ++k) {
      v2f Bk;
      Bk.x = m0[k] ? xv : 0.0f;
      Bk.y = m1[k] ? xv : 0.0f;
      D = __builtin_amdgcn_wmma_f32_16x16x4_f32(
              /*neg_a=*/false, A[k], /*neg_b=*/false, Bk,
              /*c_mod=*/(short)0, D, /*reuse_a=*/false, /*reuse_b=*/false);
    }

    // Store straight from the D layout: per VGPR v, lanes 0-15 write 64B
    // contiguous at row h0+v, lanes 16-31 write 64B at row h0+v+8.
    float* o = out + (size_t)bt * 2048 + (size_t)(f0 + n);
#pragma unroll
    for (int v = 0; v < 8; ++v)
      o[(size_t)(h0 + v + 8 * hi) * 32] = D[v];
  }
}

extern "C" void kernel_launch(void* const* d_in, const int* in_sizes, int n_in,
                              void* d_out, int out_size, void* d_ws, size_t ws_size,
                              hipStream_t stream) {
  const float* x = (const float*)d_in[0];   // [B,T,F] f32
  const float* W = (const float*)d_in[1];   // [F,H]   f32
  const float* b = (const float*)d_in[2];   // [F,H]   f32
  float* out = (float*)d_out;               // [B,T,H,F] f32

  const int num_bt = in_sizes[0] / 32;      // B*T = 32768
  int blocks = num_bt < 4096 ? num_bt : 4096;

  hipLaunchKernelGGL(fused_mul_bias_wmma, dim3(blocks), dim3(256), 0, stream,
                     x, W, b, out, num_bt);
}